// FlowWarpConsistencyLoss_27522150433005
// MI455X (gfx1250) — compile-verified
//
#include <hip/hip_runtime.h>
#include <hip/hip_bf16.h>

// ---------------------------------------------------------------------------
// FlowWarpConsistencyLoss for MI455X (gfx1250, wave32)
//
// Pass 1: grayscale(pred_next) -> d_ws   (16 MB, L2-resident gather target)
// Pass 2: stream pred_prev (float4), upsample flow, 4-tap gather, partials
// Reduce: per-wave V_WMMA_F32_16X16X4_F32 vs all-ones B  +  shfl_xor(16)
//         -> 2 global f32 atomics per wave
// Final:  loss = sum / max(count, 1); out[0] = out[1] = loss
// ---------------------------------------------------------------------------

typedef __attribute__((ext_vector_type(2))) float v2f;
typedef __attribute__((ext_vector_type(8))) float v8f;

#define IMG_H   512
#define IMG_W   512
#define IMG_N   (IMG_H * IMG_W)          // 262144 pixels / image
#define IMG_CN  (3 * IMG_N)              // 786432 floats / image (3ch)
#define FLOW_HW 128
#define FLOW_N  (FLOW_HW * FLOW_HW)      // 16384
#define FLOW_BN (2 * FLOW_N)             // 32768 floats / image (2ch)
#define NBATCH  16
#define GRAY_FLOATS (NBATCH * IMG_N)     // 4,194,304 floats = 16 MB scratch
#define EPS2    (0.001f * 0.001f)
#define FLOW_SC (127.0f / 511.0f)        // align-corners src scale 512->128

// Wave-wide (32-lane) sum using the f32 WMMA unit.
// A: each lane contributes {partial, 0}; B: all ones.
// D[m][n] = sum_k A[m][k]  ==> every column of D sums to the wave total.
// Lanes n and n^16 jointly hold column n of D (documented C/D layout), so
// per-lane sum of the 8 D registers + one xor-16 shuffle = wave total.
__device__ __forceinline__ float wave_sum_wmma(float partial) {
    v2f a;  a.x = partial; a.y = 0.0f;
    v2f b1; b1.x = 1.0f;   b1.y = 1.0f;
    v8f c = {};
    c = __builtin_amdgcn_wmma_f32_16x16x4_f32(
            /*neg_a=*/false, a, /*neg_b=*/false, b1,
            /*c_mod=*/(short)0, c, /*reuse_a=*/false, /*reuse_b=*/false);
    float t = c[0] + c[1] + c[2] + c[3] + c[4] + c[5] + c[6] + c[7];
    t += __shfl_xor(t, 16, 32);
    return t;
}

__global__ void fw_init(float* __restrict__ acc) {
    acc[0] = 0.0f;   // loss sum
    acc[1] = 0.0f;   // valid count
}

// Pass 1: grayscale pred_next. 4 pixels / thread, float4 per channel.
// 4096 blocks * 256 threads * 4 px = 16 * 262144 px exactly.
__global__ void fw_gray_next(const float* __restrict__ next,
                             float* __restrict__ gray) {
    int t  = blockIdx.x * blockDim.x + threadIdx.x;
    int b  = t >> 16;                 // 65536 threads per image
    int p  = (t & 65535) << 2;        // 4 consecutive pixels
    const float* base = next + b * IMG_CN + p;
    float4 r  = *(const float4*)(base);
    float4 g  = *(const float4*)(base + IMG_N);
    float4 bl = *(const float4*)(base + 2 * IMG_N);
    float4 o;
    o.x = 0.299f * r.x + 0.587f * g.x + 0.114f * bl.x;
    o.y = 0.299f * r.y + 0.587f * g.y + 0.114f * bl.y;
    o.z = 0.299f * r.z + 0.587f * g.z + 0.114f * bl.z;
    o.w = 0.299f * r.w + 0.587f * g.w + 0.114f * bl.w;
    *(float4*)(gray + b * IMG_N + p) = o;
}

__device__ __forceinline__ float flow_lerp(const float* __restrict__ f,
                                           int fy0, int fy1, int fx0, int fx1,
                                           float wx, float wy) {
    float v00 = f[fy0 * FLOW_HW + fx0];
    float v01 = f[fy0 * FLOW_HW + fx1];
    float v10 = f[fy1 * FLOW_HW + fx0];
    float v11 = f[fy1 * FLOW_HW + fx1];
    float top = v00 + wx * (v01 - v00);
    float bot = v10 + wx * (v11 - v10);
    return top + wy * (bot - top);
}

// Pass 2: warp + robust-L1 + masked accumulation, WMMA wave reduction.
__global__ void fw_main(const float* __restrict__ prev,
                        const float* __restrict__ flow,
                        const float* __restrict__ gray,
                        float* __restrict__ acc) {
    int t = blockIdx.x * blockDim.x + threadIdx.x;
    int b = t >> 16;
    int p = (t & 65535) << 2;         // 4 consecutive pixels, same row
    int y = p >> 9;
    int x = p & 511;

    const float* pb = prev + b * IMG_CN + p;
    float4 r  = *(const float4*)(pb);
    float4 g  = *(const float4*)(pb + IMG_N);
    float4 bl = *(const float4*)(pb + 2 * IMG_N);
    float pg[4];
    pg[0] = 0.299f * r.x + 0.587f * g.x + 0.114f * bl.x;
    pg[1] = 0.299f * r.y + 0.587f * g.y + 0.114f * bl.y;
    pg[2] = 0.299f * r.z + 0.587f * g.z + 0.114f * bl.z;
    pg[3] = 0.299f * r.w + 0.587f * g.w + 0.114f * bl.w;

    const float* gb = gray + b * IMG_N;
    const float* fu = flow + b * FLOW_BN;
    const float* fv = fu + FLOW_N;

    // Row-constant flow-grid coords (align-corners)
    float syf = (float)y * FLOW_SC;
    int   fy0 = (int)syf;
    float wyf = syf - (float)fy0;
    int   fy1 = min(fy0 + 1, FLOW_HW - 1);

    float loss_acc = 0.0f, cnt_acc = 0.0f;

    #pragma unroll
    for (int i = 0; i < 4; ++i) {
        float xf  = (float)(x + i);
        float sxf = xf * FLOW_SC;
        int   fx0 = (int)sxf;
        float wxf = sxf - (float)fx0;
        int   fx1 = min(fx0 + 1, FLOW_HW - 1);

        float u = flow_lerp(fu, fy0, fy1, fx0, fx1, wxf, wyf) * 4.0f;
        float v = flow_lerp(fv, fy0, fy1, fx0, fx1, wxf, wyf) * 4.0f;

        float gx = xf + u;
        float gy = (float)y + v;
        float valid = (gx >= 0.0f && gx <= 511.0f &&
                       gy >= 0.0f && gy <= 511.0f) ? 1.0f : 0.0f;

        float x0f = floorf(gx), y0f = floorf(gy);
        float wx  = gx - x0f,   wy  = gy - y0f;
        int   ix0 = (int)x0f,   iy0 = (int)y0f;
        int   ix1 = ix0 + 1,    iy1 = iy0 + 1;

        // branchless per-corner zero-padded fetch (keeps EXEC uniform)
        float okx0 = (ix0 >= 0 && ix0 <= 511) ? 1.0f : 0.0f;
        float okx1 = (ix1 >= 0 && ix1 <= 511) ? 1.0f : 0.0f;
        float oky0 = (iy0 >= 0 && iy0 <= 511) ? 1.0f : 0.0f;
        float oky1 = (iy1 >= 0 && iy1 <= 511) ? 1.0f : 0.0f;
        int cx0 = min(max(ix0, 0), 511), cx1 = min(max(ix1, 0), 511);
        int cy0 = min(max(iy0, 0), 511), cy1 = min(max(iy1, 0), 511);

        float v00 = oky0 * okx0 * gb[cy0 * IMG_W + cx0];
        float v01 = oky0 * okx1 * gb[cy0 * IMG_W + cx1];
        float v10 = oky1 * okx0 * gb[cy1 * IMG_W + cx0];
        float v11 = oky1 * okx1 * gb[cy1 * IMG_W + cx1];

        float warped = v00 * (1.0f - wx) * (1.0f - wy)
                     + v01 * wx          * (1.0f - wy)
                     + v10 * (1.0f - wx) * wy
                     + v11 * wx          * wy;

        float d = pg[i] - warped;
        loss_acc += sqrtf(d * d + EPS2) * valid;
        cnt_acc  += valid;
    }

    // WMMA wave reduction (EXEC is all-1s: no divergent branch above)
    float wl = wave_sum_wmma(loss_acc);
    float wc = wave_sum_wmma(cnt_acc);
    if ((threadIdx.x & 31) == 0) {
        atomicAdd(&acc[0], wl);
        atomicAdd(&acc[1], wc);
    }
}

__global__ void fw_finalize(const float* __restrict__ acc,
                            float* __restrict__ out) {
    float s = acc[0];
    float c = fmaxf(acc[1], 1.0f);
    float loss = s / c;
    out[0] = loss;   // FLOW_WARP_WEIGHT (=1.0) * loss
    out[1] = loss;
}

extern "C" void kernel_launch(void* const* d_in, const int* in_sizes, int n_in,
                              void* d_out, int out_size, void* d_ws, size_t ws_size,
                              hipStream_t stream) {
    (void)in_sizes; (void)n_in; (void)out_size; (void)ws_size;
    const float* prev = (const float*)d_in[0];   // (16,3,512,512) f32
    const float* next = (const float*)d_in[1];   // (16,3,512,512) f32
    const float* flow = (const float*)d_in[2];   // (16,2,128,128) f32
    float* out  = (float*)d_out;                 // 2 f32 scalars
    float* gray = (float*)d_ws;                  // 16 MB gray(next)
    float* acc  = gray + GRAY_FLOATS;            // 2 f32 accumulators

    fw_init<<<1, 1, 0, stream>>>(acc);
    fw_gray_next<<<4096, 256, 0, stream>>>(next, gray);
    fw_main<<<4096, 256, 0, stream>>>(prev, flow, gray, acc);
    fw_finalize<<<1, 1, 0, stream>>>(acc, out);
}